// Divergence_Gradient_symm_65661460021676
// MI455X (gfx1250) — compile-verified
//
#include <hip/hip_runtime.h>

// Divergence(avg(M) * grad(mu)) with periodic boundaries, B=16, C=1, H=W=1024.
// Memory-bound stencil: optimized for the MI455X data-movement path:
//  - GLOBAL_LOAD_ASYNC_TO_LDS_B128 staging (ASYNCcnt), s_wait_asynccnt
//  - LDS tile with halo rows, wrap handled at load time
//  - non-temporal b128 stores so the output stream does not evict the
//    input working set (~128MB ~= global L2) from L2.

typedef float f4 __attribute__((ext_vector_type(4)));

#define WDIM   1024
#define HDIM   1024
#define NBATCH 16
#define TILE_H 4
#define TROWS  (TILE_H + 2)

__device__ __forceinline__ void async_ld_b128(unsigned lds_off, const float* gptr) {
  unsigned long long ga = (unsigned long long)(size_t)gptr;
  asm volatile("global_load_async_to_lds_b128 %0, %1, off"
               :: "v"(lds_off), "v"(ga)
               : "memory");
}

__device__ __forceinline__ void wait_async0() {
  asm volatile("s_wait_asynccnt 0x0" ::: "memory");
}

__global__ __launch_bounds__(256) void Divergence_Gradient_symm_65661460021676_kernel(
    const float* __restrict__ Mg, const float* __restrict__ mug, float* __restrict__ out)
{
  __shared__ float sM[TROWS][WDIM];
  __shared__ float sU[TROWS][WDIM];

  const int tid = threadIdx.x;          // 0..255
  const int x0  = tid << 2;             // 0..1020, 16B aligned
  const int tilesPerImg = HDIM / TILE_H;
  const int bi = blockIdx.x / tilesPerImg;
  const int y0 = (blockIdx.x - bi * tilesPerImg) * TILE_H;
  const size_t imgBase = (size_t)bi * (size_t)(HDIM * WDIM);

  // ---- Stage TROWS rows (y0-1 .. y0+TILE_H, wrapped) of M and mu into LDS ----
#pragma unroll
  for (int j = 0; j < TROWS; ++j) {
    const int gy = (y0 - 1 + j + HDIM) & (HDIM - 1);
    const size_t rowBase = imgBase + (size_t)gy * WDIM + (size_t)x0;
    async_ld_b128((unsigned)(size_t)&sM[j][x0], Mg  + rowBase);
    async_ld_b128((unsigned)(size_t)&sU[j][x0], mug + rowBase);
  }
  wait_async0();       // this wave's async->LDS transfers landed
  __syncthreads();     // all waves' transfers landed

  const int xm1 = (x0 - 1) & (WDIM - 1);   // wrapped left neighbor column
  const int xp4 = (x0 + 4) & (WDIM - 1);   // wrapped right neighbor column

#pragma unroll
  for (int j = 1; j <= TILE_H; ++j) {
    const float* mC = sM[j];
    const float* uC = sU[j];

    const f4 mc  = *(const f4*)&mC[x0];
    const f4 uc  = *(const f4*)&uC[x0];
    const float ml = mC[xm1], ul = uC[xm1];
    const float mr = mC[xp4], ur = uC[xp4];

    const f4 mUp = *(const f4*)&sM[j - 1][x0];
    const f4 uUp = *(const f4*)&sU[j - 1][x0];
    const f4 mDn = *(const f4*)&sM[j + 1][x0];
    const f4 uDn = *(const f4*)&sU[j + 1][x0];

    // W-direction face fluxes at faces x0-1 .. x0+3 (5 faces, shared)
    const float m_[6] = {ml, mc.x, mc.y, mc.z, mc.w, mr};
    const float u_[6] = {ul, uc.x, uc.y, uc.z, uc.w, ur};
    float vw[5];
#pragma unroll
    for (int k = 0; k < 5; ++k)
      vw[k] = 0.5f * (m_[k + 1] + m_[k]) * (u_[k + 1] - u_[k]);

    f4 o;
    o.x = vw[1] - vw[0];
    o.y = vw[2] - vw[1];
    o.z = vw[3] - vw[2];
    o.w = vw[4] - vw[3];

    // H-direction: vH(y,x) - vH(y-1,x)
#define HTERM(c)                                                    \
    o.c += 0.5f * (mDn.c + mc.c) * (uDn.c - uc.c)                   \
         - 0.5f * (mc.c + mUp.c) * (uc.c - uUp.c);
    HTERM(x) HTERM(y) HTERM(z) HTERM(w)
#undef HTERM

    const size_t orow = imgBase + (size_t)(y0 + j - 1) * WDIM + (size_t)x0;
    __builtin_nontemporal_store(o, (f4*)(out + orow));
  }
}

extern "C" void kernel_launch(void* const* d_in, const int* in_sizes, int n_in,
                              void* d_out, int out_size, void* d_ws, size_t ws_size,
                              hipStream_t stream) {
  (void)in_sizes; (void)n_in; (void)out_size; (void)d_ws; (void)ws_size;
  const float* Mg  = (const float*)d_in[0];   // "M"
  const float* mug = (const float*)d_in[1];   // "mu"
  float* out = (float*)d_out;

  const dim3 grid(NBATCH * (HDIM / TILE_H));  // 4096 blocks
  const dim3 block(256);                      // 8 wave32s
  Divergence_Gradient_symm_65661460021676_kernel<<<grid, block, 0, stream>>>(Mg, mug, out);
}